// GGNNLayer_85882166051572
// MI455X (gfx1250) — compile-verified
//
#include <hip/hip_runtime.h>
#include <hip/hip_bf16.h>

#define HID 128
#define NTYP 6
#define LDSP 132   // padded LDS row stride (floats) -> conflict-free column reads

typedef float v2f __attribute__((ext_vector_type(2)));
typedef float v8f __attribute__((ext_vector_type(8)));

__device__ __forceinline__ v8f wmma_f32_16x16x4(v2f a, v2f b, v8f c) {
    // 8 args: (neg_a, A, neg_b, B, c_mod, C, reuse_a, reuse_b)
    return __builtin_amdgcn_wmma_f32_16x16x4_f32(
        false, a, false, b, (short)0, c, false, false);
}

__device__ __forceinline__ float sigmoidf_(float x) {
    return 1.0f / (1.0f + __expf(-x));
}

// ---------------------------------------------------------------------------
// proj_t = X @ W_edge[:, t*H:(t+1)*H] + b_edge[t*H:(t+1)*H]
// block = 256 threads = 8 waves; block covers 32 rows x 128 cols.
// Each wave owns 16 cols and TWO 16-row sub-tiles -> every B fragment feeds
// two v_wmma_f32_16x16x4_f32 ops (halves weight traffic per FLOP).
// ---------------------------------------------------------------------------
__global__ __launch_bounds__(256) void proj_gemm_kernel(
    const float* __restrict__ X, const float* __restrict__ W_edge,
    const float* __restrict__ b_edge, float* __restrict__ proj,
    int Nn, int t)
{
    __shared__ float xs[32 * LDSP];
    const int tid = threadIdx.x;
    const int m0  = blockIdx.x * 32;

    // warm L2/L0 with this type's weight columns (128 rows x 512B slices)
    if (tid < HID)
        __builtin_prefetch(W_edge + (size_t)tid * (NTYP * HID) + t * HID, 0, 1);

    for (int i = tid; i < 32 * HID; i += 256) {
        int r = i >> 7, c = i & 127;
        int gr = m0 + r; if (gr >= Nn) gr = Nn - 1;
        xs[r * LDSP + c] = X[(size_t)gr * HID + c];
    }
    __syncthreads();

    const int wave = tid >> 5;
    const int lane = tid & 31;
    const int hi   = lane >> 4;       // K sub-pair select / D row-half select
    const int ln   = lane & 15;
    const int col  = wave * 16 + ln;  // 0..127

    const float bias = b_edge[t * HID + col];
    v8f acc0, acc1;
    #pragma unroll
    for (int v = 0; v < 8; ++v) { acc0[v] = bias; acc1[v] = bias; }

    const float* Wcol = W_edge + (size_t)t * HID + col;  // element [k][col] at k*(T*H)

    #pragma unroll
    for (int k0 = 0; k0 < HID; k0 += 4) {
        const int ka = k0 + 2 * hi;
        v2f b, a0, a1;
        b.x  = Wcol[(size_t)ka * (NTYP * HID)];
        b.y  = Wcol[(size_t)(ka + 1) * (NTYP * HID)];
        a0.x = xs[ln * LDSP + ka];
        a0.y = xs[ln * LDSP + ka + 1];
        a1.x = xs[(ln + 16) * LDSP + ka];
        a1.y = xs[(ln + 16) * LDSP + ka + 1];
        acc0 = wmma_f32_16x16x4(a0, b, acc0);
        acc1 = wmma_f32_16x16x4(a1, b, acc1);
    }

    #pragma unroll
    for (int v = 0; v < 8; ++v) {
        const int gr0 = m0 + v + 8 * hi;
        const int gr1 = gr0 + 16;
        if (gr0 < Nn) proj[(size_t)gr0 * HID + col] = acc0[v];
        if (gr1 < Nn) proj[(size_t)gr1 * HID + col] = acc1[v];
    }
}

// ---------------------------------------------------------------------------
// For edges of type t: agg[dest] += proj[src]   (one wave per edge, float4 gather)
// ---------------------------------------------------------------------------
__global__ __launch_bounds__(256) void edge_scatter_kernel(
    const float* __restrict__ proj, const int* __restrict__ src,
    const int* __restrict__ dst, const int* __restrict__ types,
    const int* __restrict__ num_edges_ptr, float* __restrict__ agg,
    int E, int t)
{
    const int ne   = *num_edges_ptr;
    const int wave = threadIdx.x >> 5;
    const int lane = threadIdx.x & 31;
    const int e    = blockIdx.x * 8 + wave;
    if (e >= E || e >= ne) return;
    if (types[e] != t) return;

    const int s = src[e];
    const int d = dst[e];
    const float4 v = ((const float4*)(proj + (size_t)s * HID))[lane];
    float* a = agg + (size_t)d * HID + lane * 4;
    atomicAdd(a + 0, v.x);
    atomicAdd(a + 1, v.y);
    atomicAdd(a + 2, v.z);
    atomicAdd(a + 3, v.w);
}

// ---------------------------------------------------------------------------
// Fused GRU: h = agg (lives in `out` on entry), x = node_embeddings
// r = sig(xWir+bir+hWhr); z = sig(xWiz+biz+hWhz)
// n = tanh(xWin+bin + r*(hWhn+bhn)); out = (1-z)*n + z*h
// block covers 32 rows x 128 cols; each wave: 16 cols, 2 M sub-tiles,
// 8 fp32 WMMA accumulators, B fragments shared across sub-tiles.
// ---------------------------------------------------------------------------
__global__ __launch_bounds__(256) void gru_kernel(
    const float* __restrict__ X,
    const float* __restrict__ Wir, const float* __restrict__ bir,
    const float* __restrict__ Wiz, const float* __restrict__ biz,
    const float* __restrict__ Win, const float* __restrict__ bin,
    const float* __restrict__ Whr, const float* __restrict__ Whz,
    const float* __restrict__ Whn, const float* __restrict__ bhn,
    float* __restrict__ out, int Nn)
{
    __shared__ float xs[32 * LDSP];
    __shared__ float hs[32 * LDSP];
    const int tid = threadIdx.x;
    const int m0  = blockIdx.x * 32;

    // warm caches with the six 64KB weight matrices (256B per thread each)
    {
        const size_t off = (size_t)tid * 64;   // 256 threads * 64 floats = 16K floats
        __builtin_prefetch(Wir + off, 0, 1);
        __builtin_prefetch(Wiz + off, 0, 1);
        __builtin_prefetch(Win + off, 0, 1);
        __builtin_prefetch(Whr + off, 0, 1);
        __builtin_prefetch(Whz + off, 0, 1);
        __builtin_prefetch(Whn + off, 0, 1);
    }

    for (int i = tid; i < 32 * HID; i += 256) {
        int r = i >> 7, c = i & 127;
        int gr = m0 + r; if (gr >= Nn) gr = Nn - 1;
        xs[r * LDSP + c] = X[(size_t)gr * HID + c];
        hs[r * LDSP + c] = out[(size_t)gr * HID + c];
    }
    __syncthreads();

    const int wave = tid >> 5;
    const int lane = tid & 31;
    const int hi   = lane >> 4;
    const int ln   = lane & 15;
    const int col  = wave * 16 + ln;

    v8f accr0, accz0, accnx0, accnh0;
    v8f accr1, accz1, accnx1, accnh1;
    {
        const float br = bir[col], bz = biz[col], bn = bin[col], bh = bhn[col];
        #pragma unroll
        for (int v = 0; v < 8; ++v) {
            accr0[v] = br; accz0[v] = bz; accnx0[v] = bn; accnh0[v] = bh;
            accr1[v] = br; accz1[v] = bz; accnx1[v] = bn; accnh1[v] = bh;
        }
    }

    #pragma unroll
    for (int k0 = 0; k0 < HID; k0 += 4) {
        const int ka = k0 + 2 * hi;
        v2f ax0, ax1, ah0, ah1;
        ax0.x = xs[ln * LDSP + ka];        ax0.y = xs[ln * LDSP + ka + 1];
        ax1.x = xs[(ln + 16) * LDSP + ka]; ax1.y = xs[(ln + 16) * LDSP + ka + 1];
        ah0.x = hs[ln * LDSP + ka];        ah0.y = hs[ln * LDSP + ka + 1];
        ah1.x = hs[(ln + 16) * LDSP + ka]; ah1.y = hs[(ln + 16) * LDSP + ka + 1];

        const size_t o0 = (size_t)ka * HID + col;
        const size_t o1 = (size_t)(ka + 1) * HID + col;
        v2f bir_f, biz_f, bin_f, bhr_f, bhz_f, bhn_f;
        bir_f.x = Wir[o0]; bir_f.y = Wir[o1];
        biz_f.x = Wiz[o0]; biz_f.y = Wiz[o1];
        bin_f.x = Win[o0]; bin_f.y = Win[o1];
        bhr_f.x = Whr[o0]; bhr_f.y = Whr[o1];
        bhz_f.x = Whz[o0]; bhz_f.y = Whz[o1];
        bhn_f.x = Whn[o0]; bhn_f.y = Whn[o1];

        accr0  = wmma_f32_16x16x4(ax0, bir_f, accr0);
        accr0  = wmma_f32_16x16x4(ah0, bhr_f, accr0);
        accr1  = wmma_f32_16x16x4(ax1, bir_f, accr1);
        accr1  = wmma_f32_16x16x4(ah1, bhr_f, accr1);
        accz0  = wmma_f32_16x16x4(ax0, biz_f, accz0);
        accz0  = wmma_f32_16x16x4(ah0, bhz_f, accz0);
        accz1  = wmma_f32_16x16x4(ax1, biz_f, accz1);
        accz1  = wmma_f32_16x16x4(ah1, bhz_f, accz1);
        accnx0 = wmma_f32_16x16x4(ax0, bin_f, accnx0);
        accnx1 = wmma_f32_16x16x4(ax1, bin_f, accnx1);
        accnh0 = wmma_f32_16x16x4(ah0, bhn_f, accnh0);
        accnh1 = wmma_f32_16x16x4(ah1, bhn_f, accnh1);
    }

    #pragma unroll
    for (int v = 0; v < 8; ++v) {
        const int m0r = v + 8 * hi;
        const int gr0 = m0 + m0r;
        const int gr1 = gr0 + 16;
        if (gr0 < Nn) {
            const float hval = hs[m0r * LDSP + col];
            const float r = sigmoidf_(accr0[v]);
            const float z = sigmoidf_(accz0[v]);
            const float n = tanhf(accnx0[v] + r * accnh0[v]);
            out[(size_t)gr0 * HID + col] = (1.0f - z) * n + z * hval;
        }
        if (gr1 < Nn) {
            const float hval = hs[(m0r + 16) * LDSP + col];
            const float r = sigmoidf_(accr1[v]);
            const float z = sigmoidf_(accz1[v]);
            const float n = tanhf(accnx1[v] + r * accnh1[v]);
            out[(size_t)gr1 * HID + col] = (1.0f - z) * n + z * hval;
        }
    }
}

// ---------------------------------------------------------------------------
extern "C" void kernel_launch(void* const* d_in, const int* in_sizes, int n_in,
                              void* d_out, int out_size, void* d_ws, size_t ws_size,
                              hipStream_t stream) {
    const float* X      = (const float*)d_in[0];
    const float* W_edge = (const float*)d_in[1];
    const float* b_edge = (const float*)d_in[2];
    const float* W_ir   = (const float*)d_in[3];
    const float* b_ir   = (const float*)d_in[4];
    const float* W_iz   = (const float*)d_in[5];
    const float* b_iz   = (const float*)d_in[6];
    const float* W_in   = (const float*)d_in[7];
    const float* b_in   = (const float*)d_in[8];
    const float* W_hr   = (const float*)d_in[9];
    const float* W_hz   = (const float*)d_in[10];
    const float* W_hn   = (const float*)d_in[11];
    const float* b_hn   = (const float*)d_in[12];
    const int*   srci   = (const int*)d_in[13];
    const int*   dsti   = (const int*)d_in[14];
    const int*   etyp   = (const int*)d_in[15];
    const int*   nep    = (const int*)d_in[16];

    const int Nn = in_sizes[0] / HID;
    const int E  = in_sizes[13];
    float* out  = (float*)d_out;
    float* proj = (float*)d_ws;           // needs Nn*HID floats (51.2 MB)

    const int mtiles  = (Nn + 31) / 32;
    const int eblocks = (E + 7) / 8;

    // d_out doubles as agg buffer
    hipMemsetAsync(d_out, 0, (size_t)Nn * HID * sizeof(float), stream);

    for (int t = 0; t < NTYP; ++t) {
        proj_gemm_kernel<<<mtiles, 256, 0, stream>>>(X, W_edge, b_edge, proj, Nn, t);
        edge_scatter_kernel<<<eblocks, 256, 0, stream>>>(proj, srci, dsti, etyp,
                                                         nep, out, E, t);
    }

    gru_kernel<<<mtiles, 256, 0, stream>>>(X, W_ir, b_ir, W_iz, b_iz, W_in, b_in,
                                           W_hr, W_hz, W_hn, b_hn, out, Nn);
}